// PyMLSTMLayer_72567767433568
// MI455X (gfx1250) — compile-verified
//
#include <hip/hip_runtime.h>
#include <math.h>

// mLSTM single step, H = 4096, fp32.
// d_in: 0:x 1:h_prev(unused) 2:C_prev 3:n_prev 4:Wq 5:Wk 6:Wv 7:Wo
//       8:bq 9:bk 10:bv 11:bo 12:wi 13:bi 14:wf 15:bf
// d_out: [h (H)] [C (H*H)] [n (H)]

#define H 4096

typedef __attribute__((ext_vector_type(2))) float v2f;
typedef __attribute__((ext_vector_type(8))) float v8f;

__device__ __forceinline__ float wave_reduce_sum32(float v) {
#pragma unroll
  for (int off = 16; off >= 1; off >>= 1) v += __shfl_xor(v, off, 32);
  return v;
}

// ---------------------------------------------------------------------------
// Kernel 1: scalar gates  i = exp(wi.x + bi), f = sigmoid(wf.x + bf)
// ---------------------------------------------------------------------------
__global__ __launch_bounds__(256) void gates_kernel(
    const float* __restrict__ x, const float* __restrict__ wi,
    const float* __restrict__ bi, const float* __restrict__ wf,
    const float* __restrict__ bf, float* __restrict__ ws) {
  __shared__ float si[8], sf[8];
  const int tid = threadIdx.x;
  float ai = 0.f, af = 0.f;
  for (int j = tid; j < H; j += 256) {
    const float xv = x[j];
    ai = fmaf(wi[j], xv, ai);
    af = fmaf(wf[j], xv, af);
  }
  ai = wave_reduce_sum32(ai);
  af = wave_reduce_sum32(af);
  const int wave = tid >> 5, lane = tid & 31;
  if (lane == 0) { si[wave] = ai; sf[wave] = af; }
  __syncthreads();
  if (tid == 0) {
    float ti = 0.f, tf = 0.f;
#pragma unroll
    for (int w = 0; w < 8; ++w) { ti += si[w]; tf += sf[w]; }
    ws[0] = expf(ti + bi[0]);                        // i_gate
    ws[1] = 1.0f / (1.0f + expf(-(tf + bf[0])));     // f_gate
  }
}

// ---------------------------------------------------------------------------
// Kernel 2: q,k,v,og mat-vecs. One wave per output row; x staged in LDS.
// 16384 rows total (4 matrices x 4096). Fully coalesced 512B/wave/iter reads.
// ---------------------------------------------------------------------------
__global__ __launch_bounds__(256) void qkvo_matvec_kernel(
    const float* __restrict__ x,
    const float* __restrict__ Wq, const float* __restrict__ Wk,
    const float* __restrict__ Wv, const float* __restrict__ Wo,
    const float* __restrict__ bq, const float* __restrict__ bk,
    const float* __restrict__ bv, const float* __restrict__ bo,
    float* __restrict__ ws) {
  __shared__ float xs[H];
  const int tid = threadIdx.x;
  float4* xs4 = (float4*)xs;
  const float4* x4 = (const float4*)x;
  for (int i = tid; i < H / 4; i += 256) xs4[i] = x4[i];
  __syncthreads();

  const int wave = tid >> 5, lane = tid & 31;
  const int rg = blockIdx.x * 8 + wave;   // 0..16383
  const int mat = rg >> 12;               // 0..3
  const int row = rg & (H - 1);

  const float* W = (mat == 0) ? Wq : (mat == 1) ? Wk : (mat == 2) ? Wv : Wo;
  const float* b = (mat == 0) ? bq : (mat == 1) ? bk : (mat == 2) ? bv : bo;
  const float scale = (mat == 1) ? (1.0f / 64.0f) : 1.0f;   // 1/sqrt(4096)
  float* dst = ws + 16 + (size_t)mat * H;                   // q,k,v,og packed

  const float4* Wrow = (const float4*)(W + (size_t)row * H);
  float acc = 0.f;
#pragma unroll 4
  for (int j = lane; j < H / 4; j += 32) {
    const float4 wv = Wrow[j];
    const float4 xv = xs4[j];
    acc = fmaf(wv.x, xv.x, acc);
    acc = fmaf(wv.y, xv.y, acc);
    acc = fmaf(wv.z, xv.z, acc);
    acc = fmaf(wv.w, xv.w, acc);
  }
  acc = wave_reduce_sum32(acc);
  if (lane == 0) dst[row] = (acc + b[row]) * scale;
}

// ---------------------------------------------------------------------------
// Kernel 3: C_new = f*C_prev + i*v k^T via V_WMMA_F32_16X16X4_F32 (K=1 real,
// K=1..3 zero padded), fused with y = C_prev @ q partial sums.
// One block = one 16-row strip; 8 waves stride over 256 column tiles.
// C_prev/C_out are one-pass streams -> non-temporal (TH=NT) to keep the
// 192MB L2 available for the weight stream and the hot q/k/v vectors.
// ---------------------------------------------------------------------------
__global__ __launch_bounds__(256) void update_kernel(
    const float* __restrict__ Cprev, const float* __restrict__ ws,
    float* __restrict__ Cout, float* __restrict__ ywr) {
  __shared__ float ylds[8 * 16];
  const float ig = ws[0], fg = ws[1];
  const float* q = ws + 16;
  const float* k = ws + 16 + H;
  const float* v = ws + 16 + 2 * H;

  const int tid = threadIdx.x;
  const int wave = tid >> 5, lane = tid & 31;
  const int half = lane >> 4;         // 0: rows m0..m0+7 | 1: rows m0+8..15
  const int lcol = lane & 15;         // tile column
  const int m0 = blockIdx.x * 16;
  const int rowbase = m0 + half * 8;

  // A operand (16x4 f32): only column K=0 nonzero = i_gate * v[m0..m0+15].
  // Load in all lanes (no EXEC divergence), select with cndmask.
  const float va = ig * v[m0 + (lane & 15)];
  v2f A;
  A.x = (half == 0) ? va : 0.0f;
  A.y = 0.0f;

  float yacc[8];
#pragma unroll
  for (int r = 0; r < 8; ++r) yacc[r] = 0.f;

  for (int ct = wave; ct < H / 16; ct += 8) {   // uniform trip count: EXEC full
    const int c0 = ct * 16;
    const int col = c0 + lcol;
    // B operand (4x16 f32): only row K=0 nonzero = k[c0..c0+15].
    const float kb = k[col];          // all lanes load (halves dup: L2/LDS hit)
    v2f B;
    B.x = (half == 0) ? kb : 0.0f;
    B.y = 0.0f;

    const float qc = q[col];
    const float* cp = Cprev + (size_t)rowbase * H + col;
    v8f cin;
    float raw[8];
#pragma unroll
    for (int r = 0; r < 8; ++r)
      raw[r] = __builtin_nontemporal_load(cp + (size_t)r * H);
#pragma unroll
    for (int r = 0; r < 8; ++r) {
      cin[r] = fg * raw[r];                    // f * C_prev tile
      yacc[r] = fmaf(raw[r], qc, yacc[r]);     // fused C_prev @ q
    }
    // D = (i*v) k^T + f*C_prev
    v8f d = __builtin_amdgcn_wmma_f32_16x16x4_f32(
        false, A, false, B, (short)0, cin, false, false);
    float* co = Cout + (size_t)rowbase * H + col;
#pragma unroll
    for (int r = 0; r < 8; ++r)
      __builtin_nontemporal_store(d[r], co + (size_t)r * H);
  }

  // reduce y partials across the 16 lanes of each half-wave
#pragma unroll
  for (int r = 0; r < 8; ++r) {
    float t = yacc[r];
    t += __shfl_xor(t, 1, 32);
    t += __shfl_xor(t, 2, 32);
    t += __shfl_xor(t, 4, 32);
    t += __shfl_xor(t, 8, 32);
    yacc[r] = t;
  }
  if (lcol == 0) {
#pragma unroll
    for (int r = 0; r < 8; ++r) ylds[wave * 16 + half * 8 + r] = yacc[r];
  }
  __syncthreads();
  if (tid < 16) {
    float s = 0.f;
#pragma unroll
    for (int w = 0; w < 8; ++w) s += ylds[w * 16 + tid];
    ywr[m0 + tid] = s;   // y[m] = C_prev[m,:] . q
  }
}

// ---------------------------------------------------------------------------
// Kernel 4: n = f*n_prev + i*k ; denom = max(|n.q|,1);
//           h = sigmoid(og) * (f*y + i*(k.q)*v) / denom
// ---------------------------------------------------------------------------
__global__ __launch_bounds__(512) void finalize_kernel(
    const float* __restrict__ nprev, const float* __restrict__ ws,
    float* __restrict__ h_out, float* __restrict__ n_out) {
  __shared__ float skq[16], snq[16], bc[2];
  const int tid = threadIdx.x;
  const float ig = ws[0], fg = ws[1];
  const float* q  = ws + 16;
  const float* k  = ws + 16 + H;
  const float* v  = ws + 16 + 2 * H;
  const float* og = ws + 16 + 3 * H;
  const float* y  = ws + 16 + 4 * H;

  float kq = 0.f, nq = 0.f;
  for (int j = tid; j < H; j += 512) {
    const float qj = q[j], kj = k[j];
    const float nj = fmaf(fg, nprev[j], ig * kj);
    n_out[j] = nj;
    kq = fmaf(kj, qj, kq);
    nq = fmaf(nj, qj, nq);
  }
  kq = wave_reduce_sum32(kq);
  nq = wave_reduce_sum32(nq);
  const int wave = tid >> 5, lane = tid & 31;
  if (lane == 0) { skq[wave] = kq; snq[wave] = nq; }
  __syncthreads();
  if (tid == 0) {
    float a = 0.f, b = 0.f;
#pragma unroll
    for (int w = 0; w < 16; ++w) { a += skq[w]; b += snq[w]; }
    bc[0] = a;
    bc[1] = fmaxf(fabsf(b), 1.0f);
  }
  __syncthreads();
  const float kq_t = bc[0], denom = bc[1];
  for (int j = tid; j < H; j += 512) {
    const float ht = (fmaf(fg, y[j], ig * kq_t * v[j])) / denom;
    const float o = 1.0f / (1.0f + expf(-og[j]));
    h_out[j] = o * ht;
  }
}

// ---------------------------------------------------------------------------
extern "C" void kernel_launch(void* const* d_in, const int* in_sizes, int n_in,
                              void* d_out, int out_size, void* d_ws, size_t ws_size,
                              hipStream_t stream) {
  const float* x     = (const float*)d_in[0];
  const float* Cprev = (const float*)d_in[2];
  const float* nprev = (const float*)d_in[3];
  const float* Wq    = (const float*)d_in[4];
  const float* Wk    = (const float*)d_in[5];
  const float* Wv    = (const float*)d_in[6];
  const float* Wo    = (const float*)d_in[7];
  const float* bq    = (const float*)d_in[8];
  const float* bk    = (const float*)d_in[9];
  const float* bv    = (const float*)d_in[10];
  const float* bo    = (const float*)d_in[11];
  const float* wi    = (const float*)d_in[12];
  const float* bi    = (const float*)d_in[13];
  const float* wf    = (const float*)d_in[14];
  const float* bf    = (const float*)d_in[15];

  float* h_out = (float*)d_out;
  float* C_out = h_out + H;
  float* n_out = C_out + (size_t)H * H;

  float* wsf = (float*)d_ws;   // [0]:i [1]:f | +16: q,k,v,og,y (H each) ~82KB
  float* ywr = wsf + 16 + 4 * (size_t)H;

  gates_kernel<<<1, 256, 0, stream>>>(x, wi, bi, wf, bf, wsf);
  qkvo_matvec_kernel<<<(4 * H) / 8, 256, 0, stream>>>(
      x, Wq, Wk, Wv, Wo, bq, bk, bv, bo, wsf);
  update_kernel<<<H / 16, 256, 0, stream>>>(Cprev, wsf, C_out, ywr);
  finalize_kernel<<<1, 512, 0, stream>>>(nprev, wsf, h_out, n_out);
}